// RelAttFusion_11201274708208
// MI455X (gfx1250) — compile-verified
//
#include <hip/hip_runtime.h>

// ---------------------------------------------------------------------------
// RelAttFusion on gfx1250 (CDNA5, wave32) using V_WMMA_F32_16X16X4_F32.
// B=2, S=2048, D=512, H=8, DH=64, DS=2.  All math in fp32 to match reference.
//
// Pipeline (11 launches, all batched over (b,h) where applicable):
//   1-4. Q/K/V/rel projections           (WMMA GEMM, A.B^T + bias)
//   5.   content = q.k^T  for all bh     (WMMA GEMM, batched grid.y)
//   6.   pos     = shift(q.rel^T)        (WMMA GEMM with fused skew-store)
//   7.   q_ds/k_ds scale                 (elementwise, batched)
//   8.   cs = (q_ds.k_ds^T)/DS           (WMMA GEMM, batched)
//   9.   softmax(score) -> attn          (fused row softmax, batched)
//  10.   out_h = attn.v                  (WMMA GEMM A.B, batched)
//  11.   out = O.Wo^T + bo               (WMMA GEMM)
// ---------------------------------------------------------------------------

#define Bn   2
#define Sn   2048
#define Dn   512
#define Hn   8
#define DHn  64
#define DSn  2
#define Rn   (2 * Sn - 1)   // 4095
#define SDn  (Sn / DSn)     // 1024

typedef __attribute__((ext_vector_type(2))) float v2f;
typedef __attribute__((ext_vector_type(8))) float v8f;

__device__ __forceinline__ v8f wmma4(v2f a, v2f b, v8f c) {
  // 8 args: (neg_a, A, neg_b, B, c_mod, C, reuse_a, reuse_b)
  return __builtin_amdgcn_wmma_f32_16x16x4_f32(
      false, a, false, b, (short)0, c, false, false);
}

// ---------------------------------------------------------------------------
// Register-blocked wave-level WMMA GEMM: each wave computes a 32x64 block
// (2 M-subtiles x 4 N-subtiles, 8 v8f accumulators) so A fragments are
// reused 4x and B fragments 2x  ->  1.5 floats/lane per WMMA.
// Per k-step ALL 6 fragment loads are issued before the 8-WMMA burst so the
// compiler can keep multiple loads in flight (staged s_wait_loadcnt).
//
//   TRANSB = true :  C = alpha * A[M,K] * Bm[N,K]^T + bias[N]
//   TRANSB = false:  C = alpha * A[M,K] * Bm[K,N]   + bias[N]
//   SHIFT  = true :  instead of C[m,n], scatter to C[m, n-(M-1)+m] when that
//                    column lies in [0,M)  (Music-Transformer skew); tiles
//                    entirely outside the band exit early (wave-uniform).
//
// K must be a multiple of 4.  M,N arbitrary: loads clamped (EXEC stays all-1s
// for WMMA), stores masked.  blockIdx.y selects (b,h) via the six strides.
// ---------------------------------------------------------------------------
template <bool TRANSB, bool SHIFT>
__global__ __launch_bounds__(128) void gemm_wmma(
    const float* __restrict__ A, int lda,
    const float* __restrict__ Bm, int ldb,
    float* __restrict__ C, int ldc,
    int M, int N, int K,
    const float* __restrict__ bias, float alpha,
    size_t sAb, size_t sAh, size_t sBb, size_t sBh, size_t sCb, size_t sCh)
{
  const int bh = blockIdx.y;
  const int bb = bh / Hn;
  const int hh = bh % Hn;
  A  += (size_t)bb * sAb + (size_t)hh * sAh;
  Bm += (size_t)bb * sBb + (size_t)hh * sBh;
  C  += (size_t)bb * sCb + (size_t)hh * sCh;

  const int mt = (M + 31) >> 5;   // 32-row blocks
  const int nt = (N + 63) >> 6;   // 64-col blocks
  const int tile = blockIdx.x * blockDim.y + threadIdx.y;
  if (tile >= mt * nt) return;                  // wave-uniform exit
  const int tm = tile / nt;
  const int tn = tile % nt;

  if (SHIFT) {
    // band check: t = n - (M-1) + m must intersect [0, M)
    const int tmin = tn * 64 - (M - 1) + tm * 32;
    const int tmax = tn * 64 + 63 - (M - 1) + tm * 32 + 31;
    if (tmax < 0 || tmin >= M) return;          // wave-uniform exit
  }

  const int lane = threadIdx.x;                 // 0..31 (wave32)
  const int lp   = lane & 15;
  const int hl   = lane >> 4;                   // lane-half selects K pair

  const int m0  = tm * 32 + lp;
  const int mr0 = (m0 < M) ? m0 : (M - 1);      // clamp: keep EXEC full
  const int m1  = m0 + 16;
  const int mr1 = (m1 < M) ? m1 : (M - 1);
  const float* __restrict__ arow0 = A + (size_t)mr0 * lda;
  const float* __restrict__ arow1 = A + (size_t)mr1 * lda;

  int nn[4], nr[4];
  const float* brow[4];
#pragma unroll
  for (int j = 0; j < 4; ++j) {
    nn[j] = tn * 64 + j * 16 + lp;
    nr[j] = (nn[j] < N) ? nn[j] : (N - 1);
    brow[j] = TRANSB ? (Bm + (size_t)nr[j] * ldb) : Bm;
  }

  v8f acc[8];
#pragma unroll
  for (int i = 0; i < 8; ++i) acc[i] = (v8f){0.f,0.f,0.f,0.f,0.f,0.f,0.f,0.f};

  for (int k = 0; k < K; k += 4) {
    const int kb = k + hl * 2;
    // ---- issue all 6 fragment loads up front (memory-level parallelism) ----
    v2f a0, a1, bf[4];
    a0.x = arow0[kb]; a0.y = arow0[kb + 1];
    a1.x = arow1[kb]; a1.y = arow1[kb + 1];
#pragma unroll
    for (int j = 0; j < 4; ++j) {
      if (TRANSB) {
        bf[j].x = brow[j][kb];
        bf[j].y = brow[j][kb + 1];
      } else {
        bf[j].x = Bm[(size_t)kb * ldb + nr[j]];
        bf[j].y = Bm[(size_t)(kb + 1) * ldb + nr[j]];
      }
    }
    // ---- 8-WMMA burst ----
#pragma unroll
    for (int j = 0; j < 4; ++j) {
      acc[j]     = wmma4(a0, bf[j], acc[j]);
      acc[4 + j] = wmma4(a1, bf[j], acc[4 + j]);
    }
  }

  // C/D layout: VGPR v holds row (v + 8*hl) of the 16x16 subtile, col lane&15.
#pragma unroll
  for (int mi = 0; mi < 2; ++mi) {
#pragma unroll
    for (int j = 0; j < 4; ++j) {
      const int n = nn[j];
      if (n >= N) continue;
      const float badd = (!SHIFT && bias) ? bias[n] : 0.f;
#pragma unroll
      for (int v = 0; v < 8; ++v) {
        const int cm = tm * 32 + mi * 16 + v + hl * 8;
        if (cm >= M) continue;
        const float val = alpha * acc[mi * 4 + j][v] + badd;
        if (SHIFT) {
          const int t = n - (M - 1) + cm;       // skewed destination column
          if (t >= 0 && t < M) C[(size_t)cm * ldc + t] = val;
        } else {
          C[(size_t)cm * ldc + n] = val;
        }
      }
    }
  }
}

// ---------------------------------------------------------------------------
// Depthwise stride-DS "conv" == strided slice + per-channel scale/bias.
// Batched over all (b,h): q_ds/k_ds stored as [bh, S/DS, DH].
// ---------------------------------------------------------------------------
__global__ __launch_bounds__(256) void ds_scale_kernel(
    const float* __restrict__ Q, const float* __restrict__ Kb,
    const float* __restrict__ qw, const float* __restrict__ qb,
    const float* __restrict__ kw, const float* __restrict__ kbias,
    float* __restrict__ qds, float* __restrict__ kds)
{
  const int total = Bn * Hn * SDn * DHn;
  const int idx = blockIdx.x * blockDim.x + threadIdx.x;
  if (idx >= total) return;
  const int bh  = idx / (SDn * DHn);
  const int rem = idx - bh * (SDn * DHn);
  const int sd  = rem / DHn;
  const int d   = rem - sd * DHn;
  const int b   = bh / Hn;
  const int h   = bh % Hn;
  const size_t src = ((size_t)b * Sn + (size_t)sd * DSn) * Dn + h * DHn + d;
  qds[idx] = Q[src] * qw[d] + qb[d];
  kds[idx] = Kb[src] * kw[d] + kbias[d];
}

// ---------------------------------------------------------------------------
// Fused score assembly + row softmax.  grid = (S, B*H); 256 thr = 8 waves.
//   score = (content + w_fuse * cs[s/2, t/2] + pos) / sqrt(D)
// ---------------------------------------------------------------------------
__global__ __launch_bounds__(256) void softmax_kernel(
    const float* __restrict__ content, const float* __restrict__ pos,
    const float* __restrict__ csb, const float* __restrict__ wfuse,
    float* __restrict__ attn)
{
  __shared__ float red[8];
  const int s  = blockIdx.x;
  const int bh = blockIdx.y;
  const size_t scr_off = (size_t)bh * Sn * Sn + (size_t)s * Sn;
  const float* __restrict__ crow  = content + scr_off;
  const float* __restrict__ prow  = pos + scr_off;
  const float* __restrict__ csrow = csb + (size_t)bh * SDn * SDn
                                        + (size_t)(s >> 1) * SDn;
  const float wf = wfuse[0];
  const float inv_scale = 0.04419417382415922f;  // 1/sqrt(512)

  const int t0   = threadIdx.x;
  const int lane = threadIdx.x & 31;
  const int wid  = threadIdx.x >> 5;

  float sc[8];
  float mx = -INFINITY;
#pragma unroll
  for (int i = 0; i < 8; ++i) {
    const int t = t0 + i * 256;
    const float v = (crow[t] + wf * csrow[t >> 1] + prow[t]) * inv_scale;
    sc[i] = v;
    mx = fmaxf(mx, v);
  }
  for (int off = 16; off; off >>= 1) mx = fmaxf(mx, __shfl_down(mx, off, 32));
  if (lane == 0) red[wid] = mx;
  __syncthreads();
  if (wid == 0) {
    float m = (lane < 8) ? red[lane] : -INFINITY;
    for (int off = 4; off; off >>= 1) m = fmaxf(m, __shfl_down(m, off, 32));
    if (lane == 0) red[0] = m;
  }
  __syncthreads();
  mx = red[0];

  float sum = 0.f;
#pragma unroll
  for (int i = 0; i < 8; ++i) {
    sc[i] = expf(sc[i] - mx);
    sum += sc[i];
  }
  for (int off = 16; off; off >>= 1) sum += __shfl_down(sum, off, 32);
  __syncthreads();                    // everyone done reading red[0]
  if (lane == 0) red[wid] = sum;
  __syncthreads();
  if (wid == 0) {
    float m = (lane < 8) ? red[lane] : 0.f;
    for (int off = 4; off; off >>= 1) m += __shfl_down(m, off, 32);
    if (lane == 0) red[0] = m;
  }
  __syncthreads();
  const float rinv = 1.0f / red[0];

  float* __restrict__ arow = attn + scr_off;
#pragma unroll
  for (int i = 0; i < 8; ++i) arow[t0 + i * 256] = sc[i] * rinv;
}

// ---------------------------------------------------------------------------
// Host-side launchers
// ---------------------------------------------------------------------------
template <bool TRANSB, bool SHIFT>
static void launch_gemm(const float* A, int lda, const float* B, int ldb,
                        float* C, int ldc, int M, int N, int K,
                        const float* bias, float alpha, int batch,
                        size_t sAb, size_t sAh, size_t sBb, size_t sBh,
                        size_t sCb, size_t sCh, hipStream_t st) {
  const int tiles = ((M + 31) / 32) * ((N + 63) / 64);
  dim3 blk(32, 4);
  dim3 grd((tiles + 3) / 4, batch);
  gemm_wmma<TRANSB, SHIFT><<<grd, blk, 0, st>>>(
      A, lda, B, ldb, C, ldc, M, N, K, bias, alpha,
      sAb, sAh, sBb, sBh, sCb, sCh);
}

extern "C" void kernel_launch(void* const* d_in, const int* in_sizes, int n_in,
                              void* d_out, int out_size, void* d_ws, size_t ws_size,
                              hipStream_t stream) {
  const float* query   = (const float*)d_in[0];
  const float* key     = (const float*)d_in[1];
  const float* value   = (const float*)d_in[2];
  const float* pos_emb = (const float*)d_in[3];
  const float* Wq = (const float*)d_in[4];
  const float* bq = (const float*)d_in[5];
  const float* Wk = (const float*)d_in[6];
  const float* bk = (const float*)d_in[7];
  const float* Wv = (const float*)d_in[8];
  const float* bv = (const float*)d_in[9];
  const float* Wo = (const float*)d_in[10];
  const float* bo = (const float*)d_in[11];
  const float* Wp = (const float*)d_in[12];
  const float* bp = (const float*)d_in[13];
  const float* qds_w = (const float*)d_in[14];
  const float* qds_b = (const float*)d_in[15];
  const float* kds_w = (const float*)d_in[16];
  const float* kds_b = (const float*)d_in[17];
  const float* w_fuse = (const float*)d_in[18];

  float* out = (float*)d_out;

  // d_out layout: out | attn | content_score | pos_score (flat, return order)
  const size_t SZ_OUT = (size_t)Bn * Sn * Dn;          // 2,097,152
  const size_t SZ_SCR = (size_t)Bn * Hn * Sn * Sn;     // 67,108,864
  float* out_o  = out;
  float* attn_o = out + SZ_OUT;
  float* cont_o = attn_o + SZ_SCR;
  float* pos_o  = cont_o + SZ_SCR;

  // workspace carve-up (floats), ~110 MB total
  float* Q   = (float*)d_ws;                           // [B*S, D]
  float* Kb  = Q   + (size_t)Bn * Sn * Dn;             // [B*S, D]
  float* V   = Kb  + (size_t)Bn * Sn * Dn;             // [B*S, D]
  float* O   = V   + (size_t)Bn * Sn * Dn;             // [B*S, D] (pre-Wo)
  float* REL = O   + (size_t)Bn * Sn * Dn;             // [4096, 64] (4095 used)
  float* QDS = REL + (size_t)4096 * DHn;               // [BH, S/DS, DH]
  float* KDS = QDS + (size_t)Bn * Hn * SDn * DHn;      // [BH, S/DS, DH]
  float* CS  = KDS + (size_t)Bn * Hn * SDn * DHn;      // [BH, S/DS, S/DS]

  const size_t sQb = (size_t)Sn * Dn, sQh = DHn;       // head strides in Q/K/V/O
  const size_t sSb = (size_t)Hn * Sn * Sn, sSh = (size_t)Sn * Sn;  // score strides
  const size_t sDSbh = (size_t)SDn * DHn;              // q_ds/k_ds per-bh stride
  const size_t sCSbh = (size_t)SDn * SDn;              // cs per-bh stride
  const int BH = Bn * Hn;

  // ---- projections: X @ W^T + b ----
  launch_gemm<true, false>(query, Dn, Wq, Dn, Q,  Dn, Bn * Sn, Dn, Dn, bq, 1.0f,
                           1, 0, 0, 0, 0, 0, 0, stream);
  launch_gemm<true, false>(key,   Dn, Wk, Dn, Kb, Dn, Bn * Sn, Dn, Dn, bk, 1.0f,
                           1, 0, 0, 0, 0, 0, 0, stream);
  launch_gemm<true, false>(value, Dn, Wv, Dn, V,  Dn, Bn * Sn, Dn, Dn, bv, 1.0f,
                           1, 0, 0, 0, 0, 0, 0, stream);
  launch_gemm<true, false>(pos_emb, DHn, Wp, DHn, REL, DHn, Rn, DHn, DHn, bp, 1.0f,
                           1, 0, 0, 0, 0, 0, 0, stream);

  // ---- content = q . k^T  (all bh at once) ----
  launch_gemm<true, false>(Q, Dn, Kb, Dn, cont_o, Sn, Sn, Sn, DHn, nullptr, 1.0f,
                           BH, sQb, sQh, sQb, sQh, sSb, sSh, stream);

  // ---- pos = skew(q . rel^T)  (fused skew-store, band tiles only) ----
  launch_gemm<true, true>(Q, Dn, REL, DHn, pos_o, Sn, Sn, Rn, DHn, nullptr, 1.0f,
                          BH, sQb, sQh, 0, 0, sSb, sSh, stream);

  // ---- downsample + per-channel scale/bias (all bh) ----
  {
    const int total = Bn * Hn * SDn * DHn;
    ds_scale_kernel<<<(total + 255) / 256, 256, 0, stream>>>(
        Q, Kb, qds_w, qds_b, kds_w, kds_b, QDS, KDS);
  }

  // ---- cs = (q_ds . k_ds^T) / DS  (all bh) ----
  launch_gemm<true, false>(QDS, DHn, KDS, DHn, CS, SDn, SDn, SDn, DHn,
                           nullptr, 1.0f / DSn,
                           BH, sDSbh * Hn, sDSbh, sDSbh * Hn, sDSbh,
                           sCSbh * Hn, sCSbh, stream);

  // ---- fused score + softmax -> attn  (all bh) ----
  {
    dim3 grd(Sn, BH);
    softmax_kernel<<<grd, 256, 0, stream>>>(cont_o, pos_o, CS, w_fuse, attn_o);
  }

  // ---- out_h = attn . v  (all bh) ----
  launch_gemm<false, false>(attn_o, Sn, V, Dn, O, Dn, Sn, DHn, Sn, nullptr, 1.0f,
                            BH, sSb, sSh, sQb, sQh, sQb, sQh, stream);

  // ---- output projection: O @ Wo^T + bo ----
  launch_gemm<true, false>(O, Dn, Wo, Dn, out_o, Dn, Bn * Sn, Dn, Dn, bo, 1.0f,
                           1, 0, 0, 0, 0, 0, 0, stream);

  (void)in_sizes; (void)n_in; (void)out_size; (void)ws_size;
}